// QuadtreeAttention_33182917329362
// MI455X (gfx1250) — compile-verified
//
#include <hip/hip_runtime.h>
#include <hip/hip_bf16.h>

// ---------------- problem constants ----------------
#define B_   2
#define NH_  8
#define HH_  64
#define WW_  64
#define C_   256
#define DH_  32
#define N_   (HH_*WW_)     // 4096
#define SCALE_ 0.17677669529663687f  // 32^-0.5

typedef __attribute__((ext_vector_type(2))) float v2f;
typedef __attribute__((ext_vector_type(8))) float v8f;

static __device__ __forceinline__ v8f wmma_f32(v2f a, v2f b, v8f c) {
    // V_WMMA_F32_16X16X4_F32 : D = A(16x4) * B(4x16) + C(16x16), fp32
    return __builtin_amdgcn_wmma_f32_16x16x4_f32(
        /*neg_a=*/false, a, /*neg_b=*/false, b,
        /*c_mod=*/(short)0, c, /*reuse_a=*/false, /*reuse_b=*/false);
}

// =====================================================================
// GEMM: out = x @ w^T ; x (M x 256), w (256 x 256) row-major.
// One wave computes one 16x16 tile with a K-loop of 64 WMMA(16x16x4).
// qkv variant writes head layout (B, NH, 4096, 32).
// =====================================================================
__global__ __launch_bounds__(256) void qta_gemm_qkv(const float* __restrict__ x,
                                                    const float* __restrict__ w,
                                                    float* __restrict__ out) {
    const int wave = threadIdx.x >> 5;
    const int lane = threadIdx.x & 31;
    const int lm = lane & 15, hf = lane >> 4;
    const int m0 = blockIdx.x * 16;                 // 512 M-tiles
    const int n0 = (blockIdx.y * 8 + wave) * 16;    // 16 N-tiles
    v8f acc = {};
    const float* xr = x + (size_t)(m0 + lm) * C_;
    const float* wr = w + (size_t)(n0 + lm) * C_;
    for (int kk = 0; kk < C_; kk += 4) {
        v2f a, b;
        a[0] = xr[kk + 0 + 2*hf];
        a[1] = xr[kk + 1 + 2*hf];
        b[0] = wr[kk + 0 + 2*hf];
        b[1] = wr[kk + 1 + 2*hf];
        acc = wmma_f32(a, b, acc);
    }
    for (int r = 0; r < 8; ++r) {
        const int m = m0 + r + 8*hf;        // global row (b*4096+s)
        const int n = n0 + lm;              // global col (h*32+d)
        const int b = m >> 12, s = m & 4095;
        const int h = n >> 5,  d = n & 31;
        out[((((size_t)b*NH_ + h)*N_) + s)*DH_ + d] = acc[r];
    }
}

__global__ __launch_bounds__(256) void qta_gemm_proj(const float* __restrict__ x,
                                                     const float* __restrict__ w,
                                                     const float* __restrict__ bias,
                                                     float* __restrict__ out) {
    const int wave = threadIdx.x >> 5;
    const int lane = threadIdx.x & 31;
    const int lm = lane & 15, hf = lane >> 4;
    const int m0 = blockIdx.x * 16;
    const int n0 = (blockIdx.y * 8 + wave) * 16;
    v8f acc = {};
    const float* xr = x + (size_t)(m0 + lm) * C_;
    const float* wr = w + (size_t)(n0 + lm) * C_;
    for (int kk = 0; kk < C_; kk += 4) {
        v2f a, b;
        a[0] = xr[kk + 0 + 2*hf];
        a[1] = xr[kk + 1 + 2*hf];
        b[0] = wr[kk + 0 + 2*hf];
        b[1] = wr[kk + 1 + 2*hf];
        acc = wmma_f32(a, b, acc);
    }
    for (int r = 0; r < 8; ++r) {
        const int m = m0 + r + 8*hf;
        const int n = n0 + lm;
        out[(size_t)m*C_ + n] = acc[r] + bias[n];
    }
}

// =====================================================================
// 2x2 average pool: in (B,NH,2h,2w,DH) -> out (B,NH,h,w,DH)
// =====================================================================
__global__ void qta_pool(const float* __restrict__ in, float* __restrict__ out,
                         int h, int w) {
    const int total = B_*NH_*h*w*DH_;
    const int i = blockIdx.x*blockDim.x + threadIdx.x;
    if (i >= total) return;
    int t = i;
    const int d = t % DH_; t /= DH_;
    const int x = t % w;   t /= w;
    const int y = t % h;   t /= h;      // t = b*NH+h
    const int W2 = 2*w;
    const float* p = in + (((size_t)t*(2*h) + 2*y)*W2 + 2*x)*DH_ + d;
    out[i] = 0.25f*(p[0] + p[DH_] + p[(size_t)W2*DH_] + p[(size_t)(W2+1)*DH_]);
}

// =====================================================================
// Level 0 (coarsest, 16x16 grid, S0=256): full attention + top-16.
// One wave per (b,h, 16-query tile).
// =====================================================================
__global__ __launch_bounds__(32) void qta_level0(const float* __restrict__ q,
                                                 const float* __restrict__ k,
                                                 const float* __restrict__ v,
                                                 float* __restrict__ msg,
                                                 float* __restrict__ ps_out,
                                                 int* __restrict__ pi_out) {
    __shared__ float S[16][256];     // softmax probs
    __shared__ float SC[16][256];    // topk scratch / m_all staging
    __shared__ float topv[16][16];
    __shared__ int   topi[16][16];

    const int blk = blockIdx.x;
    const int qt  = blk & 15;        // query tile within (b,h)
    const int bh  = blk >> 4;
    const int lane = threadIdx.x;
    const int lm = lane & 15, hf = lane >> 4;
    const int qrow0 = qt * 16;
    const float* Q = q + (size_t)bh*256*DH_;
    const float* K = k + (size_t)bh*256*DH_;
    const float* V = v + (size_t)bh*256*DH_;

    // ---- scores: 16 key tiles, K-loop over Dh=32 ----
    for (int nt = 0; nt < 16; ++nt) {
        v8f acc = {};
        const float* qr = Q + (size_t)(qrow0 + lm)*DH_;
        const float* kr = K + (size_t)(nt*16 + lm)*DH_;
        for (int kk = 0; kk < DH_; kk += 4) {
            v2f a, b;
            a[0] = SCALE_ * qr[kk + 0 + 2*hf];
            a[1] = SCALE_ * qr[kk + 1 + 2*hf];
            b[0] = kr[kk + 0 + 2*hf];
            b[1] = kr[kk + 1 + 2*hf];
            acc = wmma_f32(a, b, acc);
        }
        for (int r = 0; r < 8; ++r)
            S[r + 8*hf][nt*16 + lm] = acc[r];
    }
    __syncthreads();

    // ---- row softmax + top-16 (lane m owns row m) ----
    if (lane < 16) {
        const int m = lane;
        float mx = -1e30f;
        for (int j = 0; j < 256; ++j) mx = fmaxf(mx, S[m][j]);
        float sum = 0.f;
        for (int j = 0; j < 256; ++j) { float e = __expf(S[m][j] - mx); S[m][j] = e; sum += e; }
        const float inv = 1.f / sum;
        for (int j = 0; j < 256; ++j) { S[m][j] *= inv; SC[m][j] = S[m][j]; }
        for (int t = 0; t < 16; ++t) {
            float bv = -1e30f; int bi = 0;
            for (int j = 0; j < 256; ++j) if (SC[m][j] > bv) { bv = SC[m][j]; bi = j; }
            topv[m][t] = bv; topi[m][t] = bi; SC[m][bi] = -1e30f;
        }
    }
    __syncthreads();

    // ---- m_all = A @ V via WMMA (K=256, two d-tiles) ----
    v8f a0 = {}, a1 = {};
    for (int kk = 0; kk < 256; kk += 4) {
        v2f a, b0, b1;
        a[0] = S[lm][kk + 0 + 2*hf];
        a[1] = S[lm][kk + 1 + 2*hf];
        const float* vr0 = V + (size_t)(kk + 0 + 2*hf)*DH_;
        const float* vr1 = V + (size_t)(kk + 1 + 2*hf)*DH_;
        b0[0] = vr0[lm];      b0[1] = vr1[lm];
        b1[0] = vr0[16 + lm]; b1[1] = vr1[16 + lm];
        a0 = wmma_f32(a, b0, a0);
        a1 = wmma_f32(a, b1, a1);
    }
    for (int r = 0; r < 8; ++r) {
        SC[r + 8*hf][lm]      = a0[r];
        SC[r + 8*hf][16 + lm] = a1[r];
    }
    __syncthreads();

    // ---- msg = m_all - m_sel; write p_score/p_idx ----
    for (int m = 0; m < 16; ++m) {           // lane == d
        float msel = 0.f;
        for (int t = 0; t < 16; ++t)
            msel += topv[m][t] * V[(size_t)topi[m][t]*DH_ + lane];
        msg[((size_t)bh*256 + qrow0 + m)*DH_ + lane] = SC[m][lane] - msel;
    }
    if (lane < 16) {
        for (int m = 0; m < 16; ++m) {
            ps_out[((size_t)bh*256 + qrow0 + m)*16 + lane] = topv[m][lane];
            pi_out[((size_t)bh*256 + qrow0 + m)*16 + lane] = topi[m][lane];
        }
    }
}

// =====================================================================
// Level t (t>=1): per parent quad, 4 child queries x 64 child keys.
// One wave per (b,h,parent). gh,gw = coarse grid; fine grid = 2gh x 2gw.
// =====================================================================
__global__ __launch_bounds__(32) void qta_levelt(const float* __restrict__ qf,
                                                 const float* __restrict__ kf,
                                                 const float* __restrict__ vf,
                                                 const float* __restrict__ ps_in,
                                                 const int*   __restrict__ pi_in,
                                                 float* __restrict__ msg,
                                                 float* __restrict__ ps_out,
                                                 int*   __restrict__ pi_out,
                                                 int gh, int gw, int is_last) {
    __shared__ float sL[4][64];    // s = p_score * softmax probs
    __shared__ float mA[4][32];    // m_all staging
    __shared__ float psh[16];
    __shared__ int   ci[64];       // gathered child key indices (k*4+c)
    __shared__ float topv[4][16];
    __shared__ int   topi[4][16];

    const int Sp = gh*gw;
    const int p  = blockIdx.x % Sp;
    const int bh = blockIdx.x / Sp;
    const int wt = 2*gw;
    const int St = 4*Sp;
    const int lane = threadIdx.x;
    const int lm = lane & 15, hf = lane >> 4;
    const float* Q = qf + (size_t)bh*St*DH_;
    const float* K = kf + (size_t)bh*St*DH_;
    const float* V = vf + (size_t)bh*St*DH_;

    if (lane < 16) {
        psh[lane] = ps_in[((size_t)bh*Sp + p)*16 + lane];
        const int pi = pi_in[((size_t)bh*Sp + p)*16 + lane];
        const int r = pi / gw, c = pi % gw;
        for (int cc = 0; cc < 4; ++cc)
            ci[lane*4 + cc] = (2*r + (cc >> 1))*wt + (2*c + (cc & 1));
    }
    __syncthreads();

    const int rp = p / gw, cp = p % gw;
    const int qm = lm & 3;                              // query row for A loads
    const int qidx = (2*rp + (qm >> 1))*wt + (2*cp + (qm & 1));
    const bool alive = (lm < 4);

    // ---- scores: 4 key tiles of 16, K-loop over Dh ----
    for (int nt = 0; nt < 4; ++nt) {
        v8f acc = {};
        const int key = ci[nt*16 + lm];
        const float* qr = Q + (size_t)qidx*DH_;
        const float* kr = K + (size_t)key*DH_;
        for (int kk = 0; kk < DH_; kk += 4) {
            v2f a, b;
            const float a0v = SCALE_ * qr[kk + 0 + 2*hf];
            const float a1v = SCALE_ * qr[kk + 1 + 2*hf];
            a[0] = alive ? a0v : 0.f;
            a[1] = alive ? a1v : 0.f;
            b[0] = kr[kk + 0 + 2*hf];
            b[1] = kr[kk + 1 + 2*hf];
            acc = wmma_f32(a, b, acc);
        }
        if (hf == 0)
            for (int r = 0; r < 4; ++r) sL[r][nt*16 + lm] = acc[r];
    }
    __syncthreads();

    // ---- per (query, parent) softmax over 4 children, scaled by p_score ----
    for (int g = lane; g < 64; g += 32) {
        const int m = g >> 4, kp = g & 15;
        float* row = &sL[m][kp*4];
        const float mx = fmaxf(fmaxf(row[0], row[1]), fmaxf(row[2], row[3]));
        const float e0 = __expf(row[0]-mx), e1 = __expf(row[1]-mx);
        const float e2 = __expf(row[2]-mx), e3 = __expf(row[3]-mx);
        const float f = psh[kp] / (e0+e1+e2+e3);
        row[0] = e0*f; row[1] = e1*f; row[2] = e2*f; row[3] = e3*f;
    }
    __syncthreads();

    // ---- m_all = s (4x64) @ vc (64x32) via WMMA ----
    for (int nt = 0; nt < 2; ++nt) {
        v8f acc = {};
        for (int kk = 0; kk < 64; kk += 4) {
            v2f a, b;
            const float a0v = sL[qm][kk + 0 + 2*hf];
            const float a1v = sL[qm][kk + 1 + 2*hf];
            a[0] = alive ? a0v : 0.f;
            a[1] = alive ? a1v : 0.f;
            b[0] = V[(size_t)ci[kk + 0 + 2*hf]*DH_ + nt*16 + lm];
            b[1] = V[(size_t)ci[kk + 1 + 2*hf]*DH_ + nt*16 + lm];
            acc = wmma_f32(a, b, acc);
        }
        if (hf == 0)
            for (int r = 0; r < 4; ++r) mA[r][nt*16 + lm] = acc[r];
    }
    __syncthreads();

    if (is_last) {
        for (int m = 0; m < 4; ++m) {       // lane == d
            const int fi = (2*rp + (m >> 1))*wt + (2*cp + (m & 1));
            msg[((size_t)bh*St + fi)*DH_ + lane] = mA[m][lane];
        }
        return;
    }

    // ---- top-16 of 64 per row (destructive on sL) ----
    if (lane < 4) {
        const int m = lane;
        for (int t = 0; t < 16; ++t) {
            float bv = -1e30f; int bi = 0;
            for (int j = 0; j < 64; ++j) if (sL[m][j] > bv) { bv = sL[m][j]; bi = j; }
            topv[m][t] = bv; topi[m][t] = bi; sL[m][bi] = -1e30f;
        }
    }
    __syncthreads();

    for (int m = 0; m < 4; ++m) {           // lane == d
        float msel = 0.f;
        for (int t = 0; t < 16; ++t)
            msel += topv[m][t] * V[(size_t)ci[topi[m][t]]*DH_ + lane];
        const int fi = (2*rp + (m >> 1))*wt + (2*cp + (m & 1));
        msg[((size_t)bh*St + fi)*DH_ + lane] = mA[m][lane] - msel;
    }
    if (lane < 16) {
        for (int m = 0; m < 4; ++m) {
            const int fi = (2*rp + (m >> 1))*wt + (2*cp + (m & 1));
            ps_out[((size_t)bh*St + fi)*16 + lane] = topv[m][lane];
            pi_out[((size_t)bh*St + fi)*16 + lane] = ci[topi[m][lane]];
        }
    }
}

// =====================================================================
// Combine: out(B,N,C) = upsampled msg0 + msg1 + msg2  (head-major -> channel)
// =====================================================================
__global__ void qta_combine(const float* __restrict__ m0, const float* __restrict__ m1,
                            const float* __restrict__ m2, float* __restrict__ out) {
    const int total = B_*N_*C_;
    const int i = blockIdx.x*blockDim.x + threadIdx.x;
    if (i >= total) return;
    const int c = i % C_;
    int t = i / C_;
    const int col = t % WW_;
    const int row = (t / WW_) % HH_;
    const int b = t / (WW_*HH_);
    const int h = c >> 5, d = c & 31;
    const int bh = b*NH_ + h;
    float val = m2[((size_t)bh*4096 + row*64 + col)*DH_ + d]
              + m1[((size_t)bh*1024 + (row >> 1)*32 + (col >> 1))*DH_ + d]
              + m0[((size_t)bh*256  + (row >> 2)*16 + (col >> 2))*DH_ + d];
    out[i] = val;
}

// =====================================================================
extern "C" void kernel_launch(void* const* d_in, const int* in_sizes, int n_in,
                              void* d_out, int out_size, void* d_ws, size_t ws_size,
                              hipStream_t stream) {
    (void)in_sizes; (void)n_in; (void)out_size; (void)ws_size;
    const float* x  = (const float*)d_in[0];
    // d_in[1]=H, d_in[2]=W (scalars; compile-time constants here)
    const float* wq = (const float*)d_in[3];
    const float* wk = (const float*)d_in[4];
    const float* wv = (const float*)d_in[5];
    const float* wp = (const float*)d_in[6];
    const float* bp = (const float*)d_in[7];

    const size_t SZ2 = (size_t)B_*NH_*4096*DH_;   // 2,097,152
    const size_t SZ1 = (size_t)B_*NH_*1024*DH_;   //   524,288
    const size_t SZ0 = (size_t)B_*NH_*256 *DH_;   //   131,072
    float* ws = (float*)d_ws; size_t o = 0;
    float* q2 = ws + o; o += SZ2;  float* k2 = ws + o; o += SZ2;  float* v2 = ws + o; o += SZ2;
    float* q1 = ws + o; o += SZ1;  float* k1 = ws + o; o += SZ1;  float* v1 = ws + o; o += SZ1;
    float* q0 = ws + o; o += SZ0;  float* k0 = ws + o; o += SZ0;  float* v0 = ws + o; o += SZ0;
    float* msg0 = ws + o; o += SZ0;
    float* msg1 = ws + o; o += SZ1;
    float* msg2 = ws + o; o += SZ2;
    float* ps0 = ws + o; o += (size_t)B_*NH_*256*16;
    int*   pi0 = (int*)(ws + o); o += (size_t)B_*NH_*256*16;
    float* ps1 = ws + o; o += (size_t)B_*NH_*1024*16;
    int*   pi1 = (int*)(ws + o); o += (size_t)B_*NH_*1024*16;
    float* attn = ws + o; o += SZ2;

    const dim3 gg(512, 2);          // 512 M-tiles x (2 blocks * 8 waves) N-tiles
    qta_gemm_qkv<<<gg, 256, 0, stream>>>(x, wq, q2);
    qta_gemm_qkv<<<gg, 256, 0, stream>>>(x, wk, k2);
    qta_gemm_qkv<<<gg, 256, 0, stream>>>(x, wv, v2);

    const int T1 = B_*NH_*32*32*DH_;
    qta_pool<<<(T1 + 255)/256, 256, 0, stream>>>(q2, q1, 32, 32);
    qta_pool<<<(T1 + 255)/256, 256, 0, stream>>>(k2, k1, 32, 32);
    qta_pool<<<(T1 + 255)/256, 256, 0, stream>>>(v2, v1, 32, 32);
    const int T0 = B_*NH_*16*16*DH_;
    qta_pool<<<(T0 + 255)/256, 256, 0, stream>>>(q1, q0, 16, 16);
    qta_pool<<<(T0 + 255)/256, 256, 0, stream>>>(k1, k0, 16, 16);
    qta_pool<<<(T0 + 255)/256, 256, 0, stream>>>(v1, v0, 16, 16);

    qta_level0<<<B_*NH_*16, 32, 0, stream>>>(q0, k0, v0, msg0, ps0, pi0);
    qta_levelt<<<B_*NH_*256,  32, 0, stream>>>(q1, k1, v1, ps0, pi0, msg1, ps1, pi1, 16, 16, 0);
    qta_levelt<<<B_*NH_*1024, 32, 0, stream>>>(q2, k2, v2, ps1, pi1, msg2, nullptr, nullptr, 32, 32, 1);

    qta_combine<<<(B_*N_*C_ + 255)/256, 256, 0, stream>>>(msg0, msg1, msg2, attn);
    qta_gemm_proj<<<gg, 256, 0, stream>>>(attn, wp, bp, (float*)d_out);
}